// GCN_72739566125755
// MI455X (gfx1250) — compile-verified
//
#include <hip/hip_runtime.h>
#include <hip/hip_bf16.h>

// ---------------------------------------------------------------------------
// 2-layer GCN for MI455X (gfx1250).
//  - Dense linears use V_WMMA_F32_16X16X4_F32 (wave32, one 16x16 output tile
//    per wave; HID/ACT == 16 == exactly one WMMA N-tile).
//  - Edge aggregation is the memory-bound part (~820MB irregular traffic):
//    one (edge,feature) pair per lane -> coalesced 64B atomic-add groups,
//    biases folded into accumulator init, self-loops folded into edge range.
// ---------------------------------------------------------------------------

typedef __attribute__((ext_vector_type(2))) float v2f;
typedef __attribute__((ext_vector_type(8))) float v8f;

#define F_IN 128
#define HID  16

// ---------------------------------------------------------------------------
// init: deg = 1 (self loop), agg1[i][f] = b1[f], out[i][f] = b2[f]
// ---------------------------------------------------------------------------
__global__ void gcn_init_kernel(float* __restrict__ deg,
                                float* __restrict__ agg1,
                                float* __restrict__ out,
                                const float* __restrict__ b1,
                                const float* __restrict__ b2,
                                int n) {
    int tid = blockIdx.x * blockDim.x + threadIdx.x;
    if (tid >= n * HID) return;
    int f = tid & (HID - 1);
    if (f == 0) deg[tid >> 4] = 1.0f;   // self-loop contribution to in-degree
    agg1[tid] = b1[f];
    out[tid]  = b2[f];
}

// ---------------------------------------------------------------------------
// degree scatter: deg[dst[e]] += 1
// ---------------------------------------------------------------------------
__global__ void gcn_deg_kernel(const int* __restrict__ dst,
                               float* __restrict__ deg, int e) {
    int i = blockIdx.x * blockDim.x + threadIdx.x;
    if (i < e) atomicAdd(&deg[dst[i]], 1.0f);
}

// deg -> rsqrt(deg) in place (deg >= 1 always; keep the reference's guard)
__global__ void gcn_rsqrt_kernel(float* __restrict__ deg, int n) {
    int i = blockIdx.x * blockDim.x + threadIdx.x;
    if (i < n) {
        float d = deg[i];
        deg[i] = d > 0.0f ? rsqrtf(d) : 0.0f;
    }
}

// ---------------------------------------------------------------------------
// GEMM  C[n x 16] = act(A[n x K]) @ B[K x 16]   via V_WMMA_F32_16X16X4_F32.
// One wave per 16-row tile; K accumulated in steps of 4.
// A-operand layout (ISA 7.12.2, 32-bit A 16x4): lane l<16 holds A[l][k..k+1],
// lane l+16 holds A[l][k+2..k+3]  -> one 8B vector load per lane per step.
// B-operand mirrors it: vgpr0/1 = rows k+kh / k+kh+1, N striped over lanes.
// C/D: vgpr r = row (r + 8*(lane>=16)), col = lane&15.
// ---------------------------------------------------------------------------
template <bool RELU, int K>
__global__ __launch_bounds__(128)
void gcn_gemm_wmma_kernel(const float* __restrict__ A,
                          const float* __restrict__ B,
                          float* __restrict__ C, int nrows) {
    const int lane = threadIdx.x & 31;
    const int wave = threadIdx.x >> 5;
    const int tile = blockIdx.x * 4 + wave;
    const int row0 = tile * 16;
    if (row0 >= nrows) return;           // wave-uniform: EXEC stays all-ones

    const int m    = lane & 15;          // output col / A row within tile
    const int half = lane >> 4;          // 0 or 1
    const int kh   = half << 1;          // K sub-offset: 0 or 2

    const float* __restrict__ arow = A + (size_t)(row0 + m) * K;

    v8f acc = {};
#pragma unroll
    for (int k = 0; k < K; k += 4) {
        v2f a = *reinterpret_cast<const v2f*>(arow + k + kh);  // 8B-aligned
        if (RELU) {
            a.x = fmaxf(a.x, 0.0f);
            a.y = fmaxf(a.y, 0.0f);
        }
        v2f b;
        b.x = B[(k + kh) * HID + m];
        b.y = B[(k + kh + 1) * HID + m];
        acc = __builtin_amdgcn_wmma_f32_16x16x4_f32(
            /*neg_a=*/false, a, /*neg_b=*/false, b,
            /*c_mod=*/(short)0, acc, /*reuse_a=*/false, /*reuse_b=*/false);
    }

    float* __restrict__ ccol = C + (size_t)row0 * HID + m;
#pragma unroll
    for (int r = 0; r < 8; ++r) {
        ccol[(size_t)(r + 8 * half) * HID] = acc[r];
    }
}

// ---------------------------------------------------------------------------
// aggregation: out[dst] += h[src] * dinv[src]*dinv[dst], over E edges plus
// N self-loops (idx >= E => src = dst = idx - E). 16 lanes per edge so each
// message is a coalesced 64B group of global_atomic_add_f32.
// ---------------------------------------------------------------------------
__global__ void gcn_agg_kernel(const float* __restrict__ h,
                               const int* __restrict__ src,
                               const int* __restrict__ dst,
                               const float* __restrict__ dinv,
                               float* __restrict__ out,
                               int e, int n) {
    int tid = blockIdx.x * blockDim.x + threadIdx.x;
    int idx = tid >> 4;
    int f   = tid & (HID - 1);
    if (idx >= e + n) return;
    int s, d;
    if (idx < e) {
        s = src[idx];
        d = dst[idx];
    } else {
        s = d = idx - e;                 // self loop
    }
    float norm = dinv[s] * dinv[d];
    atomicAdd(&out[(size_t)d * HID + f], h[(size_t)s * HID + f] * norm);
}

// ---------------------------------------------------------------------------
// launch
// ---------------------------------------------------------------------------
extern "C" void kernel_launch(void* const* d_in, const int* in_sizes, int n_in,
                              void* d_out, int out_size, void* d_ws, size_t ws_size,
                              hipStream_t stream) {
    const float* x  = (const float*)d_in[0];   // [N, 128]
    const int*   ei = (const int*)d_in[1];     // [2, E] (harness integer -> int32)
    const float* W1 = (const float*)d_in[2];   // [128, 16]
    const float* b1 = (const float*)d_in[3];   // [16]
    const float* W2 = (const float*)d_in[4];   // [16, 16]
    const float* b2 = (const float*)d_in[5];   // [16]
    float* out = (float*)d_out;                // [N, 16]

    const int N = in_sizes[0] / F_IN;          // 100000
    const int E = in_sizes[1] / 2;             // 3200000
    const int* src = ei;
    const int* dst = ei + E;

    // workspace: deg/dinv [N] | h [N*16] | agg1 [N*16]
    char* ws = (char*)d_ws;
    float* deg  = (float*)ws;
    float* hbuf = (float*)(ws + (size_t)N * sizeof(float));
    float* agg1 = (float*)(ws + (size_t)N * sizeof(float) +
                                (size_t)N * HID * sizeof(float));

    const int TB = 256;

    // 1) init accumulators (bias folded in) + self-loop degree
    gcn_init_kernel<<<(N * HID + TB - 1) / TB, TB, 0, stream>>>(
        deg, agg1, out, b1, b2, N);

    // 2) in-degree scatter, then dinv = rsqrt(deg)
    gcn_deg_kernel<<<(E + TB - 1) / TB, TB, 0, stream>>>(dst, deg, E);
    gcn_rsqrt_kernel<<<(N + TB - 1) / TB, TB, 0, stream>>>(deg, N);

    // 3) layer 1 linear: h = x @ W1   (WMMA, K=128)
    const int tiles = (N + 15) / 16;           // 6250
    const int gblk  = (tiles + 3) / 4;         // 4 waves/block
    gcn_gemm_wmma_kernel<false, F_IN><<<gblk, 128, 0, stream>>>(x, W1, hbuf, N);

    // 4) layer 1 aggregation into agg1 (pre-seeded with b1)
    const long long at = (long long)(E + N) * HID;
    gcn_agg_kernel<<<(int)((at + TB - 1) / TB), TB, 0, stream>>>(
        hbuf, src, dst, deg, agg1, E, N);

    // 5) layer 2 linear: h2 = relu(agg1) @ W2   (WMMA, K=16, ReLU fused)
    gcn_gemm_wmma_kernel<true, HID><<<gblk, 128, 0, stream>>>(agg1, W2, hbuf, N);

    // 6) layer 2 aggregation into d_out (pre-seeded with b2)
    gcn_agg_kernel<<<(int)((at + TB - 1) / TB), TB, 0, stream>>>(
        hbuf, src, dst, deg, out, E, N);
}